// QuantumGeometricAttention_27522150433470
// MI455X (gfx1250) — compile-verified
//
#include <hip/hip_runtime.h>
#include <hip/hip_bf16.h>

// Problem constants (match reference)
#define BB 2
#define SSQ 1024
#define EE 1024
#define HH 8
#define DDM 128
#define LLAY 3
#define QSCALE 0.08838834764831845f   // 1/sqrt(128)

typedef float v2f __attribute__((ext_vector_type(2)));
typedef float v8f __attribute__((ext_vector_type(8)));

static __device__ __forceinline__ v8f wmma_f32(v2f a, v2f b, v8f c) {
  // V_WMMA_F32_16X16X4_F32: D(16x16 f32) = A(16x4 f32) * B(4x16 f32) + C
  return __builtin_amdgcn_wmma_f32_16x16x4_f32(
      /*neg_a=*/false, a, /*neg_b=*/false, b,
      /*c_mod=*/(short)0, c, /*reuse_a=*/false, /*reuse_b=*/false);
}

// Prefetch distance (floats): 512B ahead of the demand stream.
#define PF_DIST 128

// ---------------------------------------------------------------------------
// Complex GEMM: C[M,N] = A[M,K] * W[N,K]^T + bias[N] (+ resid[M,N])
// All complex64 interleaved (re,im) floats. W rows contiguous (stride K).
// A row r maps to complex offset (r>>a_shift)*a_outer + (r&a_mask)*a_inner,
// which covers both uniform strides and the [B*S*H, D] view of the qkv buffer.
// Each wave computes a 16x64 tile: 4 N-subtiles reuse one A load.
// Software-pipelined: step k+4 loads issue before step k's WMMAs consume.
// ---------------------------------------------------------------------------
__global__ void __launch_bounds__(256) cgemm_wmma_kernel(
    const float* __restrict__ A, const float* __restrict__ W,
    const float* __restrict__ bias, const float* __restrict__ resid,
    float* __restrict__ C,
    int M, int N, int K,
    int a_outer, int a_inner, int a_shift, int a_mask)
{
  const int lane = threadIdx.x & 31;
  const int wave = blockIdx.x * (blockDim.x >> 5) + (threadIdx.x >> 5);
  const int ntc  = N >> 6;                 // 64-wide N chunks
  const int total = (M >> 4) * ntc;
  if (wave >= total) return;
  const int mt = wave / ntc, nt = wave - mt * ntc;
  const int m0 = mt << 4, n0 = nt << 6;

  const int lm = lane & 15;                // M row (A) / N col (B,C) within tile
  const int kh = (lane >> 4) << 1;         // K sub-offset {0,2} per ISA layout

  const int  r    = m0 + lm;
  const long arow = (long)(r >> a_shift) * a_outer + (long)(r & a_mask) * a_inner;
  const float* Ap = A + 2 * arow + 2 * kh;

  const float* Wp[4];
#pragma unroll
  for (int j = 0; j < 4; ++j)
    Wp[j] = W + 2 * ((long)(n0 + j * 16 + lm) * K + kh);

  const v8f zz = {0.f,0.f,0.f,0.f,0.f,0.f,0.f,0.f};
  v8f accR[4], accI[4];
#pragma unroll
  for (int j = 0; j < 4; ++j) { accR[j] = zz; accI[j] = zz; }

  // prologue loads (step 0)
  float4 avc = *(const float4*)(Ap);
  float4 wvc[4];
#pragma unroll
  for (int j = 0; j < 4; ++j) wvc[j] = *(const float4*)(Wp[j]);

#pragma unroll 2
  for (int k = 0; k < K - 4; k += 4) {
    // deep speculative prefetch for the HBM streaming phase
    __builtin_prefetch(Ap + 2 * k + PF_DIST, 0, 1);
#pragma unroll
    for (int j = 0; j < 4; ++j)
      __builtin_prefetch(Wp[j] + 2 * k + PF_DIST, 0, 1);

    // issue next step's loads before consuming current registers
    const float4 avn = *(const float4*)(Ap + 2 * (k + 4));
    float4 wvn[4];
#pragma unroll
    for (int j = 0; j < 4; ++j) wvn[j] = *(const float4*)(Wp[j] + 2 * (k + 4));

    v2f ar = {avc.x, avc.z};
    v2f ai = {avc.y, avc.w};
    v2f an = {-avc.y, -avc.w};             // f32 WMMA has no A-neg modifier
#pragma unroll
    for (int j = 0; j < 4; ++j) {
      v2f br = {wvc[j].x, wvc[j].z};
      v2f bi = {wvc[j].y, wvc[j].w};
      accR[j] = wmma_f32(ar, br, accR[j]);
      accR[j] = wmma_f32(an, bi, accR[j]);   // Cr = Ar*Br - Ai*Bi
      accI[j] = wmma_f32(ar, bi, accI[j]);
      accI[j] = wmma_f32(ai, br, accI[j]);   // Ci = Ar*Bi + Ai*Br
    }
    avc = avn;
#pragma unroll
    for (int j = 0; j < 4; ++j) wvc[j] = wvn[j];
  }
  {
    // epilogue step (k = K-4)
    v2f ar = {avc.x, avc.z};
    v2f ai = {avc.y, avc.w};
    v2f an = {-avc.y, -avc.w};
#pragma unroll
    for (int j = 0; j < 4; ++j) {
      v2f br = {wvc[j].x, wvc[j].z};
      v2f bi = {wvc[j].y, wvc[j].w};
      accR[j] = wmma_f32(ar, br, accR[j]);
      accR[j] = wmma_f32(an, bi, accR[j]);
      accI[j] = wmma_f32(ar, bi, accI[j]);
      accI[j] = wmma_f32(ai, br, accI[j]);
    }
  }

  // C/D 16x16 f32 layout: row = v + 8*(lane>>4), col = lane&15
  const int mrow = m0 + ((lane >> 4) << 3);
#pragma unroll
  for (int j = 0; j < 4; ++j) {
    const int n = n0 + j * 16 + lm;
    float bre = 0.f, bim = 0.f;
    if (bias) { bre = bias[2 * n]; bim = bias[2 * n + 1]; }
#pragma unroll
    for (int v = 0; v < 8; ++v) {
      const int  m   = mrow + v;
      const long off = 2 * ((long)m * N + n);
      float cr = accR[j][v] + bre;
      float ci = accI[j][v] + bim;
      if (resid) { cr += resid[off]; ci += resid[off + 1]; }
      float2 o; o.x = cr; o.y = ci;
      *(float2*)(C + off) = o;
    }
  }
}

// ---------------------------------------------------------------------------
// expmap0: per row of D complex values, v *= tanh(max(|v|,1e-6))/max(|v|,1e-6)
// One wave per row; rows are contiguous D-complex vectors ([B,S,H,D]).
// ---------------------------------------------------------------------------
__global__ void __launch_bounds__(256) expmap_kernel(float* __restrict__ T, int rows)
{
  const int row = blockIdx.x * 8 + (threadIdx.x >> 5);
  if (row >= rows) return;
  const int lane = threadIdx.x & 31;
  float2* p = (float2*)T + (long)row * DDM;

  float2 vals[4];
  float ss = 0.f;
#pragma unroll
  for (int i = 0; i < 4; ++i) {
    vals[i] = p[lane + 32 * i];
    ss += vals[i].x * vals[i].x + vals[i].y * vals[i].y;
  }
#pragma unroll
  for (int o = 16; o; o >>= 1) ss += __shfl_xor(ss, o, 32);
  float n  = sqrtf(ss);
  n        = fmaxf(n, 1e-6f);
  const float sc = tanhf(n) / n;
#pragma unroll
  for (int i = 0; i < 4; ++i) {
    vals[i].x *= sc; vals[i].y *= sc;
    p[lane + 32 * i] = vals[i];
  }
}

// ---------------------------------------------------------------------------
// scores[bh][s][t] = SCALE * Re( q[s]·conj(k[t]) ) = SCALE*(Qr·Krᵀ + Qi·Kiᵀ)
// Q,K in [B,S,H,D] complex; grid.y = b*H+h.
// ---------------------------------------------------------------------------
__global__ void __launch_bounds__(256) scores_wmma_kernel(
    const float* __restrict__ Q, const float* __restrict__ Kc,
    float* __restrict__ Sc)
{
  const int lane = threadIdx.x & 31;
  const int bh = blockIdx.y;
  const int b = bh >> 3, h = bh & 7;
  const int wave = blockIdx.x * (blockDim.x >> 5) + (threadIdx.x >> 5);
  const int ntc = SSQ >> 6;
  if (wave >= (SSQ >> 4) * ntc) return;
  const int mt = wave / ntc, nt = wave - mt * ntc;
  const int m0 = mt << 4, n0 = nt << 6;
  const int lm = lane & 15, kh = (lane >> 4) << 1;

  const float* Qp = Q + 2 * (((long)(b * SSQ + m0 + lm) * HH + h) * DDM + kh);
  const float* Kp[4];
#pragma unroll
  for (int j = 0; j < 4; ++j)
    Kp[j] = Kc + 2 * (((long)(b * SSQ + n0 + j * 16 + lm) * HH + h) * DDM + kh);

  const v8f zz = {0.f,0.f,0.f,0.f,0.f,0.f,0.f,0.f};
  v8f acc[4];
#pragma unroll
  for (int j = 0; j < 4; ++j) acc[j] = zz;

  float4 qvc = *(const float4*)(Qp);
  float4 kvc[4];
#pragma unroll
  for (int j = 0; j < 4; ++j) kvc[j] = *(const float4*)(Kp[j]);

#pragma unroll 2
  for (int k = 0; k < DDM - 4; k += 4) {
    const float4 qvn = *(const float4*)(Qp + 2 * (k + 4));
    float4 kvn[4];
#pragma unroll
    for (int j = 0; j < 4; ++j) kvn[j] = *(const float4*)(Kp[j] + 2 * (k + 4));

    v2f qr = {qvc.x, qvc.z};
    v2f qi = {qvc.y, qvc.w};
#pragma unroll
    for (int j = 0; j < 4; ++j) {
      v2f kr = {kvc[j].x, kvc[j].z};
      v2f ki = {kvc[j].y, kvc[j].w};
      acc[j] = wmma_f32(qr, kr, acc[j]);
      acc[j] = wmma_f32(qi, ki, acc[j]);    // Re(q * conj(k))
    }
    qvc = qvn;
#pragma unroll
    for (int j = 0; j < 4; ++j) kvc[j] = kvn[j];
  }
  {
    v2f qr = {qvc.x, qvc.z};
    v2f qi = {qvc.y, qvc.w};
#pragma unroll
    for (int j = 0; j < 4; ++j) {
      v2f kr = {kvc[j].x, kvc[j].z};
      v2f ki = {kvc[j].y, kvc[j].w};
      acc[j] = wmma_f32(qr, kr, acc[j]);
      acc[j] = wmma_f32(qi, ki, acc[j]);
    }
  }

  float* out = Sc + (long)bh * SSQ * SSQ;
  const int mrow = m0 + ((lane >> 4) << 3);
#pragma unroll
  for (int j = 0; j < 4; ++j)
#pragma unroll
    for (int v = 0; v < 8; ++v)
      out[(long)(mrow + v) * SSQ + n0 + j * 16 + lm] = acc[j][v] * QSCALE;
}

// ---------------------------------------------------------------------------
// Row softmax over 1024-wide rows; one wave per row.
// ---------------------------------------------------------------------------
__global__ void __launch_bounds__(256) softmax_kernel(float* __restrict__ Sc, int rows)
{
  const int row = blockIdx.x * 8 + (threadIdx.x >> 5);
  if (row >= rows) return;
  const int lane = threadIdx.x & 31;
  float* p = Sc + (long)row * SSQ;

  float mx = -3.4e38f;
  for (int i = lane; i < SSQ; i += 32) mx = fmaxf(mx, p[i]);
#pragma unroll
  for (int o = 16; o; o >>= 1) mx = fmaxf(mx, __shfl_xor(mx, o, 32));
  float sum = 0.f;
  for (int i = lane; i < SSQ; i += 32) { float e = __expf(p[i] - mx); p[i] = e; sum += e; }
#pragma unroll
  for (int o = 16; o; o >>= 1) sum += __shfl_xor(sum, o, 32);
  const float inv = 1.0f / sum;
  for (int i = lane; i < SSQ; i += 32) p[i] *= inv;
}

// ---------------------------------------------------------------------------
// O[b,s,h,:] = attn[bh][s,:] @ V[b,:,h,:]   (real attn x complex V)
// ---------------------------------------------------------------------------
__global__ void __launch_bounds__(256) av_wmma_kernel(
    const float* __restrict__ P, const float* __restrict__ V,
    float* __restrict__ O)
{
  const int lane = threadIdx.x & 31;
  const int bh = blockIdx.y;
  const int b = bh >> 3, h = bh & 7;
  const int wave = blockIdx.x * (blockDim.x >> 5) + (threadIdx.x >> 5);
  const int ntc = DDM >> 6;                // 2
  if (wave >= (SSQ >> 4) * ntc) return;
  const int mt = wave / ntc, nt = wave - mt * ntc;
  const int m0 = mt << 4, n0 = nt << 6;
  const int lm = lane & 15, kh = (lane >> 4) << 1;

  const float* Ap = P + (long)bh * SSQ * SSQ + (long)(m0 + lm) * SSQ + kh;
  const float* Vb[4];
#pragma unroll
  for (int j = 0; j < 4; ++j)
    Vb[j] = V + 2 * (((long)(b * SSQ) * HH + h) * DDM + n0 + j * 16 + lm);
  const long vstride = 2L * HH * DDM;      // floats per t step

  const v8f zz = {0.f,0.f,0.f,0.f,0.f,0.f,0.f,0.f};
  v8f accR[4], accI[4];
#pragma unroll
  for (int j = 0; j < 4; ++j) { accR[j] = zz; accI[j] = zz; }

  float2 a2c = *(const float2*)(Ap);
  float2 v0c[4], v1c[4];
#pragma unroll
  for (int j = 0; j < 4; ++j) {
    v0c[j] = *(const float2*)(Vb[j] + (long)(kh)     * vstride);
    v1c[j] = *(const float2*)(Vb[j] + (long)(kh + 1) * vstride);
  }

#pragma unroll 2
  for (int k = 0; k < SSQ - 4; k += 4) {
    // prefetch V rows ~16 t-steps ahead (8KB-strided rows; spec prefetch safe)
#pragma unroll
    for (int j = 0; j < 4; ++j)
      __builtin_prefetch(Vb[j] + (long)(k + kh + 16) * vstride, 0, 1);

    const float2 a2n = *(const float2*)(Ap + (k + 4));
    float2 v0n[4], v1n[4];
#pragma unroll
    for (int j = 0; j < 4; ++j) {
      v0n[j] = *(const float2*)(Vb[j] + (long)(k + 4 + kh)     * vstride);
      v1n[j] = *(const float2*)(Vb[j] + (long)(k + 4 + kh + 1) * vstride);
    }

    v2f a = {a2c.x, a2c.y};
#pragma unroll
    for (int j = 0; j < 4; ++j) {
      v2f vr = {v0c[j].x, v1c[j].x};
      v2f vi = {v0c[j].y, v1c[j].y};
      accR[j] = wmma_f32(a, vr, accR[j]);
      accI[j] = wmma_f32(a, vi, accI[j]);
    }
    a2c = a2n;
#pragma unroll
    for (int j = 0; j < 4; ++j) { v0c[j] = v0n[j]; v1c[j] = v1n[j]; }
  }
  {
    v2f a = {a2c.x, a2c.y};
#pragma unroll
    for (int j = 0; j < 4; ++j) {
      v2f vr = {v0c[j].x, v1c[j].x};
      v2f vi = {v0c[j].y, v1c[j].y};
      accR[j] = wmma_f32(a, vr, accR[j]);
      accI[j] = wmma_f32(a, vi, accI[j]);
    }
  }

  const int mrow = m0 + ((lane >> 4) << 3);
#pragma unroll
  for (int j = 0; j < 4; ++j)
#pragma unroll
    for (int v = 0; v < 8; ++v) {
      const int  m   = mrow + v;
      const int  d   = n0 + j * 16 + lm;
      const long off = 2 * (((long)(b * SSQ + m) * HH + h) * DDM + d);
      float2 o; o.x = accR[j][v]; o.y = accI[j][v];
      *(float2*)(O + off) = o;
    }
}

// ---------------------------------------------------------------------------
extern "C" void kernel_launch(void* const* d_in, const int* in_sizes, int n_in,
                              void* d_out, int out_size, void* d_ws, size_t ws_size,
                              hipStream_t stream)
{
  (void)in_sizes; (void)n_in; (void)out_size; (void)ws_size;

  const float* x    = (const float*)d_in[0];
  const float* Wqkv = (const float*)d_in[1];
  const float* bqkv = (const float*)d_in[2];
  const float* Wq   = (const float*)d_in[3];
  const float* bq   = (const float*)d_in[4];
  const float* Wk   = (const float*)d_in[5];
  const float* bk   = (const float*)d_in[6];
  const float* Wv   = (const float*)d_in[7];
  const float* bv   = (const float*)d_in[8];
  const float* Wout = (const float*)d_in[9];
  const float* bout = (const float*)d_in[10];
  const float* Wlay = (const float*)d_in[11];
  const float* blay = (const float*)d_in[12];
  float* outp = (float*)d_out;

  float* ws = (float*)d_ws;
  size_t o = 0;
  float* qkv  = ws + o; o += (size_t)BB * SSQ * 3 * EE * 2;  // 50.3 MB
  float* qp   = ws + o; o += (size_t)BB * SSQ * EE * 2;      // 16.8 MB each
  float* kp   = ws + o; o += (size_t)BB * SSQ * EE * 2;
  float* vp   = ws + o; o += (size_t)BB * SSQ * EE * 2;
  float* ao   = ws + o; o += (size_t)BB * SSQ * EE * 2;
  float* proj = ws + o; o += (size_t)BB * SSQ * EE * 2;
  float* bufA = ws + o; o += (size_t)BB * SSQ * EE * 2;
  float* bufB = ws + o; o += (size_t)BB * SSQ * EE * 2;
  float* sc   = ws + o; o += (size_t)BB * HH * SSQ * SSQ;    // 67 MB

  auto gemm = [&](const float* A, const float* W, const float* bias,
                  const float* resid, float* C, int M, int N, int K,
                  int aout, int ain, int ash, int amsk) {
    const int waves  = (M >> 4) * (N >> 6);
    const int blocks = (waves + 7) / 8;
    cgemm_wmma_kernel<<<dim3(blocks), dim3(256), 0, stream>>>(
        A, W, bias, resid, C, M, N, K, aout, ain, ash, amsk);
  };

  const float* cur = x;
  for (int l = 0; l < LLAY; ++l) {
    float* next = (l == 0) ? bufA : (l == 1) ? bufB : outp;

    // qkv = cur @ Wqkv^T + bqkv                       [2048 x 3072], K=1024
    gemm(cur, Wqkv, bqkv, nullptr, qkv, BB * SSQ, 3 * EE, EE, EE, 0, 0, 0);

    // per-head D x D projections on the [B*S*H, D] view of qkv
    gemm(qkv + 0 * EE, Wq, bq, nullptr, qp, BB * SSQ * HH, DDM, DDM, 3 * EE, DDM, 3, 7);
    gemm(qkv + 2 * EE, Wk, bk, nullptr, kp, BB * SSQ * HH, DDM, DDM, 3 * EE, DDM, 3, 7);
    gemm(qkv + 4 * EE, Wv, bv, nullptr, vp, BB * SSQ * HH, DDM, DDM, 3 * EE, DDM, 3, 7);

    // hyperbolic exp-map on q, k (16384 rows each)
    expmap_kernel<<<dim3(2048), dim3(256), 0, stream>>>(qp, BB * SSQ * HH);
    expmap_kernel<<<dim3(2048), dim3(256), 0, stream>>>(kp, BB * SSQ * HH);

    // scores = Re(q conj(k)) * SCALE                  [16 x 1024 x 1024]
    scores_wmma_kernel<<<dim3(128, BB * HH), dim3(256), 0, stream>>>(qp, kp, sc);

    // softmax over last dim (16384 rows of 1024)
    softmax_kernel<<<dim3(2048), dim3(256), 0, stream>>>(sc, BB * HH * SSQ);

    // out = attn @ v                                  -> [B,S,H,D] == [B,S,E]
    av_wmma_kernel<<<dim3(16, BB * HH), dim3(256), 0, stream>>>(sc, vp, ao);

    // to_out projection                               [2048 x 1024], K=1024
    gemm(ao, Wout, bout, nullptr, proj, BB * SSQ, EE, EE, EE, 0, 0, 0);

    // layer mix + fused residual: next = cur + proj @ Wl^T + bl
    gemm(proj, Wlay + (size_t)l * EE * EE * 2, blay + (size_t)l * EE * 2,
         cur, next, BB * SSQ, EE, EE, EE, 0, 0, 0);

    cur = next;
  }
}